// DotProductAttention_2465311228070
// MI455X (gfx1250) — compile-verified
//
#include <hip/hip_runtime.h>
#include <hip/hip_bf16.h>
#include <math.h>

// ---------------------------------------------------------------------------
// Sparse low-rank attention (reference: DotProductAttention_2465311228070)
// MI455X / gfx1250, wave32, WMMA f32 16x16x4 for both GEMMs.
// ---------------------------------------------------------------------------

typedef float v2f __attribute__((ext_vector_type(2)));
typedef float v8f __attribute__((ext_vector_type(8)));

#define BATCH 4
#define SEQ   2048
#define DIM   128
#define DLOW  16
#define TOPK  8
#define SCALE 0.25f          // 1/sqrt(d_low)
#define NEGBIG (-1.0e9f)
#define NEG_INF (-3.0e38f)

// ---------------------------------------------------------------------------
// Kernel 1: per-row projections q_low/k_low (fp32) and the high-rank
// correction score f[b,k] = SCALE * <Q[b,k]@Wq_high+bq, K[b,k]@Wk_high+bk>.
// One wave per row; lanes 0-15 handle the Q projections (one output column
// each), lanes 16-31 handle the K projections.
// ---------------------------------------------------------------------------
__global__ void proj_kernel(const float* __restrict__ Q, const float* __restrict__ K,
                            const float* __restrict__ WqL, const float* __restrict__ bqL,
                            const float* __restrict__ WkL, const float* __restrict__ bkL,
                            const float* __restrict__ WqH, const float* __restrict__ bqH,
                            const float* __restrict__ WkH, const float* __restrict__ bkH,
                            float* __restrict__ qlow, float* __restrict__ klow,
                            float* __restrict__ fbuf)
{
    __shared__ float qsh[8][DIM];
    __shared__ float ksh[8][DIM];
    const int lane = threadIdx.x & 31;
    const int wave = threadIdx.x >> 5;
    const int row  = blockIdx.x * 8 + wave;          // flat b*s row, 0..8191

    #pragma unroll
    for (int t = 0; t < 4; ++t) {
        qsh[wave][lane + 32 * t] = Q[(size_t)row * DIM + lane + 32 * t];
        ksh[wave][lane + 32 * t] = K[(size_t)row * DIM + lane + 32 * t];
    }
    __syncthreads();

    const int  j   = lane & 15;
    const bool isQ = lane < 16;
    const float* src = isQ ? qsh[wave] : ksh[wave];
    const float* Wl  = isQ ? WqL : WkL;
    const float* Wh  = isQ ? WqH : WkH;
    float accL = isQ ? bqL[j] : bkL[j];
    float accH = isQ ? bqH[j] : bkH[j];

    #pragma unroll 8
    for (int c = 0; c < DIM; ++c) {
        const float x = src[c];
        accL = fmaf(x, Wl[c * DLOW + j], accL);
        accH = fmaf(x, Wh[c * DLOW + j], accH);
    }
    (isQ ? qlow : klow)[(size_t)row * DLOW + j] = accL;

    // f = SCALE * sum_j qh[j]*kh[j] : qh lives in lanes 0-15, kh in 16-31.
    const float other = __shfl(accH, j + 16, 32);
    float prod = isQ ? accH * other : 0.0f;
    #pragma unroll
    for (int off = 1; off < 16; off <<= 1)
        prod += __shfl_xor(prod, off, 32);
    if (lane == 0) fbuf[row] = SCALE * prod;
}

// ---------------------------------------------------------------------------
// Kernel 2: masked low-rank scores via V_WMMA_F32_16X16X4_F32.
// S[b,q,k] = SCALE * <q_low[b,q], k_low[b,k]>  (+NEG where vl[b,k]==q)
// fp32 A 16x4 layout: lane L holds row M=L%16; VGPR0/1 = K{0,1} (lanes 0-15)
// or K{2,3} (lanes 16-31). B (= k_low^T) mirrors this, so both operands are
// contiguous float2 loads from the row-major q_low / k_low buffers.
// ---------------------------------------------------------------------------
__global__ void score_kernel(const float* __restrict__ qlow,
                             const float* __restrict__ klow,
                             const int*   __restrict__ vlens,
                             float* __restrict__ S)
{
    const int lane  = threadIdx.x;
    const int half  = lane >> 4;
    const int l16   = lane & 15;
    const int qbase = blockIdx.x * 16;
    const int kbase = blockIdx.y * 16;
    const int b     = blockIdx.z;

    const float* qrow = qlow + ((size_t)b * SEQ + qbase + l16) * DLOW;
    const float* krow = klow + ((size_t)b * SEQ + kbase + l16) * DLOW;

    v8f c = {};
    #pragma unroll
    for (int k0 = 0; k0 < DLOW; k0 += 4) {
        v2f a  = *(const v2f*)(qrow + k0 + 2 * half);
        v2f bm = *(const v2f*)(krow + k0 + 2 * half);
        c = __builtin_amdgcn_wmma_f32_16x16x4_f32(false, a, false, bm,
                                                  (short)0, c, false, false);
    }

    const int kcol = kbase + l16;
    int vl = vlens[b * SEQ + kcol];
    vl = vl < 0 ? 0 : (vl > SEQ - 1 ? SEQ - 1 : vl);

    float* Sb = S + (size_t)b * SEQ * SEQ;
    #pragma unroll
    for (int r = 0; r < 8; ++r) {
        const int q = qbase + r + 8 * half;
        float v = c[r] * SCALE;
        if (vl == q) v += NEGBIG;
        Sb[(size_t)q * SEQ + kcol] = v;
    }
}

// ---------------------------------------------------------------------------
// Kernel 3: per-row top-8 over the key axis, then scatter f[b,k] in place.
// One wave per row; each lane keeps a sorted (ascending) register top-8 of
// its 64 strided elements, then 8 rounds of butterfly argmax (low-index
// tiebreak, matching lax.top_k's ordering) pick the global top-8.
// ---------------------------------------------------------------------------
__global__ void topk_kernel(float* __restrict__ S, const float* __restrict__ fbuf)
{
    const int lane = threadIdx.x;
    const int q = blockIdx.x;
    const int b = blockIdx.y;
    float* Srow = S + ((size_t)b * SEQ + q) * SEQ;

    float val[TOPK];
    int   idx[TOPK];
    #pragma unroll
    for (int i = 0; i < TOPK; ++i) { val[i] = NEG_INF; idx[i] = -1; }

    for (int t = 0; t < SEQ / 32; ++t) {
        const int k = lane + 32 * t;
        const float v = Srow[k];
        if (v > val[0]) {
            val[0] = v; idx[0] = k;            // replace current min, bubble up
            #pragma unroll
            for (int i = 0; i < TOPK - 1; ++i) {
                if (val[i] > val[i + 1]) {
                    float tv = val[i]; val[i] = val[i + 1]; val[i + 1] = tv;
                    int   ti = idx[i]; idx[i] = idx[i + 1]; idx[i + 1] = ti;
                }
            }
        }
    }

    int chosen[TOPK];
    #pragma unroll
    for (int r = 0; r < TOPK; ++r) {
        // local max over the 8 register slots (consumed slots hold NEG_INF)
        float cv = NEG_INF; int ci = -1;
        #pragma unroll
        for (int i = 0; i < TOPK; ++i) {
            if (val[i] > cv || (val[i] == cv && idx[i] >= 0 && (ci < 0 || idx[i] < ci))) {
                cv = val[i]; ci = idx[i];
            }
        }
        // wave argmax, deterministic (value desc, then index asc)
        #pragma unroll
        for (int off = 16; off > 0; off >>= 1) {
            const float ov = __shfl_xor(cv, off, 32);
            const int   oi = __shfl_xor(ci, off, 32);
            if (ov > cv || (ov == cv && oi >= 0 && (ci < 0 || oi < ci))) { cv = ov; ci = oi; }
        }
        chosen[r] = ci;
        // owner lane retires the winning slot (global key index is unique)
        #pragma unroll
        for (int i = 0; i < TOPK; ++i)
            if (idx[i] == ci) val[i] = NEG_INF;
    }

    if (lane == 0) {
        #pragma unroll
        for (int r = 0; r < TOPK; ++r) {
            const int k = chosen[r];
            Srow[k] = fbuf[b * SEQ + k];
        }
    }
}

// ---------------------------------------------------------------------------
// Kernel 4: column-wise (query-axis) online softmax stats m[b,k], Z[b,k].
// Thread per key column -> every warp iteration reads 32 consecutive floats
// of one score row (coalesced); the 64 MiB matrix is L2-resident.
// ---------------------------------------------------------------------------
__global__ void colstats_kernel(const float* __restrict__ S,
                                float* __restrict__ mbuf, float* __restrict__ zbuf)
{
    const int k = blockIdx.x * 256 + threadIdx.x;
    const int b = blockIdx.y;
    const float* Sb = S + (size_t)b * SEQ * SEQ;
    float m = NEG_INF, z = 0.0f;
    for (int q = 0; q < SEQ; ++q) {
        const float v = Sb[(size_t)q * SEQ + k];
        const float mn = fmaxf(m, v);
        z = z * __expf(m - mn) + __expf(v - mn);
        m = mn;
    }
    mbuf[b * SEQ + k] = m;
    zbuf[b * SEQ + k] = z;
}

// ---------------------------------------------------------------------------
// Kernel 5: in-place normalize S -> attn = exp(S - m[b,k]) / Z[b,k].
// ---------------------------------------------------------------------------
__global__ void norm_kernel(float* __restrict__ S,
                            const float* __restrict__ mbuf,
                            const float* __restrict__ zbuf)
{
    const size_t i = (size_t)blockIdx.x * 256 + threadIdx.x;
    const int k = (int)(i & (SEQ - 1));
    const int b = (int)(i / ((size_t)SEQ * SEQ));
    const float v = S[i];
    S[i] = __expf(v - mbuf[b * SEQ + k]) / zbuf[b * SEQ + k];
}

// ---------------------------------------------------------------------------
// Kernel 6: out[b,q,:] = attn[b,q,:] @ V[b,:,:] via chained WMMA f32 16x16x4.
// One wave per 16(q) x 16(d) tile, 512 K-steps along the 2048 keys.
// ---------------------------------------------------------------------------
__global__ void out_kernel(const float* __restrict__ S,
                           const float* __restrict__ V,
                           float* __restrict__ out)
{
    const int lane  = threadIdx.x;
    const int half  = lane >> 4;
    const int l16   = lane & 15;
    const int qbase = blockIdx.x * 16;
    const int dbase = blockIdx.y * 16;
    const int b     = blockIdx.z;

    const float* Arow = S + ((size_t)b * SEQ + qbase + l16) * SEQ;
    const float* Vb   = V + (size_t)b * SEQ * DIM;

    v8f acc = {};
    #pragma unroll 4
    for (int k0 = 0; k0 < SEQ; k0 += 4) {
        v2f a = *(const v2f*)(Arow + k0 + 2 * half);
        v2f bv;
        bv.x = Vb[(size_t)(k0 + 2 * half)     * DIM + dbase + l16];
        bv.y = Vb[(size_t)(k0 + 2 * half + 1) * DIM + dbase + l16];
        acc = __builtin_amdgcn_wmma_f32_16x16x4_f32(false, a, false, bv,
                                                    (short)0, acc, false, false);
    }

    float* ob = out + (size_t)b * SEQ * DIM;
    #pragma unroll
    for (int r = 0; r < 8; ++r)
        ob[(size_t)(qbase + r + 8 * half) * DIM + dbase + l16] = acc[r];
}

// ---------------------------------------------------------------------------
extern "C" void kernel_launch(void* const* d_in, const int* in_sizes, int n_in,
                              void* d_out, int out_size, void* d_ws, size_t ws_size,
                              hipStream_t stream)
{
    const float* Q   = (const float*)d_in[0];
    const float* K   = (const float*)d_in[1];
    const float* V   = (const float*)d_in[2];
    const int*   vl  = (const int*)  d_in[3];
    const float* WqL = (const float*)d_in[4];
    const float* bqL = (const float*)d_in[5];
    const float* WkL = (const float*)d_in[6];
    const float* bkL = (const float*)d_in[7];
    const float* WqH = (const float*)d_in[8];
    const float* bqH = (const float*)d_in[9];
    const float* WkH = (const float*)d_in[10];
    const float* bkH = (const float*)d_in[11];
    float* out = (float*)d_out;

    // Workspace layout (floats): q_low, k_low, f, m, Z, S  (~66 MB total;
    // the 64 MiB score matrix fits in MI455X's 192 MB L2).
    float* ws   = (float*)d_ws;
    float* qlow = ws;
    float* klow = qlow + (size_t)BATCH * SEQ * DLOW;
    float* fbuf = klow + (size_t)BATCH * SEQ * DLOW;
    float* mbuf = fbuf + (size_t)BATCH * SEQ;
    float* zbuf = mbuf + (size_t)BATCH * SEQ;
    float* S    = zbuf + (size_t)BATCH * SEQ;

    proj_kernel<<<dim3(BATCH * SEQ / 8), 256, 0, stream>>>(
        Q, K, WqL, bqL, WkL, bkL, WqH, bqH, WkH, bkH, qlow, klow, fbuf);

    score_kernel<<<dim3(SEQ / 16, SEQ / 16, BATCH), 32, 0, stream>>>(qlow, klow, vl, S);

    topk_kernel<<<dim3(SEQ, BATCH), 32, 0, stream>>>(S, fbuf);

    colstats_kernel<<<dim3(SEQ / 256, BATCH), 256, 0, stream>>>(S, mbuf, zbuf);

    norm_kernel<<<dim3((unsigned)((size_t)BATCH * SEQ * SEQ / 256)), 256, 0, stream>>>(S, mbuf, zbuf);

    out_kernel<<<dim3(SEQ / 16, DIM / 16, BATCH), 32, 0, stream>>>(S, V, out);
}